// Net_79379585565572
// MI455X (gfx1250) — compile-verified
//
#include <hip/hip_runtime.h>
#include <hip/hip_bf16.h>
#include <math.h>

// Problem constants (from reference): x is (1, 4096, 4096) f32
#define D_ROWS 4096
#define N_COLS 4096
#define TPB 256
#define CPT (N_COLS / TPB)          // 16 columns per thread in the row kernel
#define ROWS_PER_CHUNK 128
#define NCHUNKS (D_ROWS / ROWS_PER_CHUNK)   // 32

typedef float v2f __attribute__((ext_vector_type(2)));
typedef float v8f __attribute__((ext_vector_type(8)));

// ---------------------------------------------------------------------------
// Wave32 sum reduction built on V_WMMA_F32_16X16X4_F32.
// A layout (ISA 7.12.2, 32-bit A 16x4): lane m (<16): VGPR0=A[m][0], VGPR1=A[m][1];
// lane m+16: VGPR0=A[m][2], VGPR1=A[m][3].
// With A-VGPR0 = p, A-VGPR1 = 0 and B = all-ones (B layout irrelevant for ones):
//   C[m][j] = p_m + p_{m+16}  for all j.
// D layout: VGPR r = C[r][lane] (lanes 0-15) / C[r+8][lane-16] (lanes 16-31),
// so per-lane sum of the 8 D VGPRs gives sum(q_0..q_7) on lanes 0-15 and
// sum(q_8..q_15) on lanes 16-31; one xor-16 shuffle completes the wave sum.
// Requires EXEC == all ones (call only from fully-converged code).
// ---------------------------------------------------------------------------
__device__ __forceinline__ float wave_sum_wmma(float p) {
    v2f A; A[0] = p;    A[1] = 0.0f;
    v2f B; B[0] = 1.0f; B[1] = 1.0f;
    v8f C = {0.0f, 0.0f, 0.0f, 0.0f, 0.0f, 0.0f, 0.0f, 0.0f};
    v8f D = __builtin_amdgcn_wmma_f32_16x16x4_f32(
        /*neg_a=*/false, A, /*neg_b=*/false, B,
        /*c_mod=*/(short)0, C, /*reuse_a=*/false, /*reuse_b=*/false);
    float t = ((D[0] + D[1]) + (D[2] + D[3])) + ((D[4] + D[5]) + (D[6] + D[7]));
    return t + __shfl_xor(t, 16, 32);
}

// ---------------------------------------------------------------------------
// Pass 1a: partial per-column min/max over chunks of 128 rows.
// grid = (N_COLS/TPB, NCHUNKS); fully coalesced; deterministic (no atomics).
// ---------------------------------------------------------------------------
__global__ __launch_bounds__(TPB)
void colminmax_partial(const float* __restrict__ x,
                       float* __restrict__ pmn, float* __restrict__ pmx) {
    const int j = blockIdx.x * TPB + threadIdx.x;     // column
    const int chunk = blockIdx.y;
    const float* p = x + (size_t)chunk * ROWS_PER_CHUNK * N_COLS + j;
    float lo = INFINITY, hi = -INFINITY;
    #pragma unroll 8
    for (int r = 0; r < ROWS_PER_CHUNK; ++r) {
        float v = p[(size_t)r * N_COLS];
        lo = fminf(lo, v);
        hi = fmaxf(hi, v);
    }
    pmn[chunk * N_COLS + j] = lo;
    pmx[chunk * N_COLS + j] = hi;
}

// ---------------------------------------------------------------------------
// Pass 1b: finish min/max, emit mn and rng (zero range -> 1, sklearn-style).
// ---------------------------------------------------------------------------
__global__ __launch_bounds__(TPB)
void colminmax_final(const float* __restrict__ pmn, const float* __restrict__ pmx,
                     float* __restrict__ mn, float* __restrict__ rng) {
    const int j = blockIdx.x * TPB + threadIdx.x;
    float lo = INFINITY, hi = -INFINITY;
    #pragma unroll
    for (int c = 0; c < NCHUNKS; ++c) {
        lo = fminf(lo, pmn[c * N_COLS + j]);
        hi = fmaxf(hi, pmx[c * N_COLS + j]);
    }
    float r = hi - lo;
    if (r == 0.0f) r = 1.0f;
    mn[j]  = lo;
    rng[j] = r;
}

// ---------------------------------------------------------------------------
// Pass 2: one workgroup per row. Row (16 KB) lives in registers (16 f32/thread);
// 3 iterations of {S,SS reduction via WMMA wave-sum + LDS cross-wave stage,
// elementwise update}; then inverse transform + store. x read exactly once.
// ---------------------------------------------------------------------------
__global__ __launch_bounds__(TPB)
void rowiter_kernel(const float* __restrict__ x,
                    const float* __restrict__ mn,
                    const float* __restrict__ rng,
                    float* __restrict__ out) {
    const int row = blockIdx.x;
    const int t   = threadIdx.x;
    const float* xr = x + (size_t)row * N_COLS;

    float a[CPT], mv[CPT], rv[CPT];

    // Load + scale: chunk c covers columns c*1024 + 4*t .. +3 (coalesced float4).
    #pragma unroll
    for (int c = 0; c < 4; ++c) {
        const int col = c * (TPB * 4) + t * 4;
        float4 xv = *reinterpret_cast<const float4*>(xr  + col);
        float4 m4 = *reinterpret_cast<const float4*>(mn  + col);
        float4 r4 = *reinterpret_cast<const float4*>(rng + col);
        mv[c*4+0] = m4.x; mv[c*4+1] = m4.y; mv[c*4+2] = m4.z; mv[c*4+3] = m4.w;
        rv[c*4+0] = r4.x; rv[c*4+1] = r4.y; rv[c*4+2] = r4.z; rv[c*4+3] = r4.w;
        a[c*4+0] = (xv.x - m4.x) / r4.x;
        a[c*4+1] = (xv.y - m4.y) / r4.y;
        a[c*4+2] = (xv.z - m4.z) / r4.z;
        a[c*4+3] = (xv.w - m4.w) / r4.w;
    }

    __shared__ float sS[TPB / 32];
    __shared__ float sSS[TPB / 32];

    const float nf      = (float)N_COLS;
    const float inv_nm1 = 1.0f / (float)(N_COLS - 1);
    const int wave = t >> 5;
    const int lane = t & 31;

    #pragma unroll
    for (int it = 0; it < 3; ++it) {
        float pS = 0.0f, pSS = 0.0f;
        #pragma unroll
        for (int k = 0; k < CPT; ++k) {
            pS  += a[k];
            pSS += a[k] * a[k];
        }
        // wave32 reduction on the matrix pipe (EXEC is all ones here)
        float wS  = wave_sum_wmma(pS);
        float wSS = wave_sum_wmma(pSS);
        if (lane == 0) { sS[wave] = wS; sSS[wave] = wSS; }
        __syncthreads();
        float S = 0.0f, SS = 0.0f;
        #pragma unroll
        for (int w = 0; w < TPB / 32; ++w) { S += sS[w]; SS += sSS[w]; }
        __syncthreads();

        const float avg = S / nf;
        const float eps = (it == 0) ? 0.005f : 0.0f;
        #pragma unroll
        for (int k = 0; k < CPT; ++k) {
            float ai   = a[k];
            float sq   = nf * ai * ai - 2.0f * ai * S + SS;
            sq         = fmaxf(sq, 0.0f);
            float rmse = sqrtf(sq * inv_nm1);
            float diff = ai - avg;
            float sg   = (diff > 0.0f) ? 1.0f : ((diff < 0.0f) ? -1.0f : 0.0f);
            a[k] = ai * (1.0f - 0.1f * rmse * sg) + eps;
        }
    }

    // inverse transform + store
    float* orow = out + (size_t)row * N_COLS;
    #pragma unroll
    for (int c = 0; c < 4; ++c) {
        const int col = c * (TPB * 4) + t * 4;
        float4 o;
        o.x = a[c*4+0] * rv[c*4+0] + mv[c*4+0];
        o.y = a[c*4+1] * rv[c*4+1] + mv[c*4+1];
        o.z = a[c*4+2] * rv[c*4+2] + mv[c*4+2];
        o.w = a[c*4+3] * rv[c*4+3] + mv[c*4+3];
        *reinterpret_cast<float4*>(orow + col) = o;
    }
}

// ---------------------------------------------------------------------------
// kernel_launch: ws layout = pmn[32*4096] | pmx[32*4096] | mn[4096] | rng[4096]
// (~1.06 MB). All launches on `stream`; no allocation / sync (graph-capture safe).
// ---------------------------------------------------------------------------
extern "C" void kernel_launch(void* const* d_in, const int* in_sizes, int n_in,
                              void* d_out, int out_size, void* d_ws, size_t ws_size,
                              hipStream_t stream) {
    (void)in_sizes; (void)n_in; (void)out_size; (void)ws_size;
    const float* x = (const float*)d_in[0];
    float* out = (float*)d_out;

    float* pmn = (float*)d_ws;
    float* pmx = pmn + (size_t)NCHUNKS * N_COLS;
    float* mn  = pmx + (size_t)NCHUNKS * N_COLS;
    float* rng = mn + N_COLS;

    dim3 gridA(N_COLS / TPB, NCHUNKS);
    colminmax_partial<<<gridA, TPB, 0, stream>>>(x, pmn, pmx);

    colminmax_final<<<N_COLS / TPB, TPB, 0, stream>>>(pmn, pmx, mn, rng);

    rowiter_kernel<<<D_ROWS, TPB, 0, stream>>>(x, mn, rng, out);
}